// SinkhornSort_5583457485535
// MI455X (gfx1250) — compile-verified
//
#include <hip/hip_runtime.h>

#define N 8192
#define TOPK 32
#define ITERS 10

typedef __attribute__((ext_vector_type(2))) float v2f;
typedef __attribute__((ext_vector_type(8))) float v8f;

// monotone float -> uint key (ascending)
__device__ __forceinline__ unsigned f2key(float f) {
  unsigned b = __float_as_uint(f);
  return b ^ ((b & 0x80000000u) ? 0xFFFFFFFFu : 0x80000000u);
}

__global__ __launch_bounds__(256) void init_v_kernel(float* v) {
  v[blockIdx.x * 256 + threadIdx.x] = 1.0f;
}

// ---------------------------------------------------------------------------
// Row pass: u[i] = 1 / sum_j exp(-d[i][j]) * v[j]
// Block: 16 rows. 8 waves split columns. WMMA f32 16x16x4 with B=ones sums
// 16x4 tiles of w = exp(-d)*v along K; C accumulates across the row.
// ---------------------------------------------------------------------------
#define RP_CHUNK 512
#define RP_STRIDE (RP_CHUNK + 4)  // pad: conflict-free ds reads (4*r banks)

__global__ __launch_bounds__(256) void row_pass(const float* __restrict__ d,
                                                const float* __restrict__ v,
                                                float* __restrict__ u) {
  __shared__ float tile[16 * RP_STRIDE];  // ~33KB
  __shared__ float red[8][16];
  const int t = threadIdx.x;
  const int wave = t >> 5;
  const int lane = t & 31;
  const int rowbase = blockIdx.x * 16;
  const int m = lane & 15;
  const int koff = (lane >> 4) << 1;

  v8f acc = {};
  v2f ones;
  ones.x = 1.0f; ones.y = 1.0f;

  for (int cc = 0; cc < N / RP_CHUNK; ++cc) {
    __syncthreads();
    // stage 16 x 512 tile of w = exp(-d)*v, coalesced b128 loads
    for (int s = 0; s < 8; ++s) {
      int idx = s * 256 + t;          // float4 index, 128 per row
      int r = idx >> 7;
      int c4 = (idx & 127) << 2;
      const float4 dd = *(const float4*)(d + (size_t)(rowbase + r) * N + cc * RP_CHUNK + c4);
      const float4 vv = *(const float4*)(v + cc * RP_CHUNK + c4);
      float4 w;
      w.x = __expf(-dd.x) * vv.x;
      w.y = __expf(-dd.y) * vv.y;
      w.z = __expf(-dd.z) * vv.z;
      w.w = __expf(-dd.w) * vv.w;
      *(float4*)&tile[r * RP_STRIDE + c4] = w;
    }
    __syncthreads();
    // wave consumes its 64-column slice: A[m][k] = tile[m][c+k], B = ones
    const int colbase = wave * 64;
    for (int kk = 0; kk < 16; ++kk) {
      int c = colbase + kk * 4;
      v2f a = *(const v2f*)&tile[m * RP_STRIDE + c + koff];
      acc = __builtin_amdgcn_wmma_f32_16x16x4_f32(false, a, false, ones,
                                                  (short)0, acc, false, false);
    }
  }
  // D[m][n] = rowsum[m] (all n). Lane with N==0 holds M = r + 8*(lane/16).
  if ((lane & 15) == 0) {
    int half = lane >> 4;
    for (int r = 0; r < 8; ++r) red[wave][half * 8 + r] = acc[r];
  }
  __syncthreads();
  if (t < 16) {
    float s = 0.0f;
    for (int w = 0; w < 8; ++w) s += red[w][t];  // fixed order: deterministic
    u[rowbase + t] = 1.0f / s;
  }
}

// ---------------------------------------------------------------------------
// Col pass stage 1: partial[rc][j] = sum_{i in chunk rc} u[i]*exp(-d[i][j])
// Grid (64 colblocks x 64 rowchunks); block covers 128 rows x 128 cols.
// WMMA with A[m][k] = w[row=k][col=m] (transposed feed), B = ones.
// ---------------------------------------------------------------------------
#define CP_STRIDE 136  // 128 + 8 pad

__global__ __launch_bounds__(256) void col_pass1(const float* __restrict__ d,
                                                 const float* __restrict__ u,
                                                 float* __restrict__ partial) {
  __shared__ float tile[16 * CP_STRIDE];
  const int t = threadIdx.x;
  const int wave = t >> 5;
  const int lane = t & 31;
  const int colbase = blockIdx.x * 128;
  const int rowbase0 = blockIdx.y * 128;
  const int m = lane & 15;
  const int koff = (lane >> 4) << 1;
  const int colw = wave * 16;

  v8f acc = {};
  v2f ones;
  ones.x = 1.0f; ones.y = 1.0f;

  for (int rchunk = 0; rchunk < 8; ++rchunk) {
    const int rowbase = rowbase0 + rchunk * 16;
    __syncthreads();
    for (int s = 0; s < 2; ++s) {
      int idx = s * 256 + t;          // float4 index, 32 per row
      int r = idx >> 5;
      int c4 = (idx & 31) << 2;
      const float4 dd = *(const float4*)(d + (size_t)(rowbase + r) * N + colbase + c4);
      const float ur = u[rowbase + r];
      float4 w;
      w.x = __expf(-dd.x) * ur;
      w.y = __expf(-dd.y) * ur;
      w.z = __expf(-dd.z) * ur;
      w.w = __expf(-dd.w) * ur;
      *(float4*)&tile[r * CP_STRIDE + c4] = w;
    }
    __syncthreads();
    for (int kk = 0; kk < 4; ++kk) {
      v2f a;
      a.x = tile[(kk * 4 + koff) * CP_STRIDE + colw + m];
      a.y = tile[(kk * 4 + koff + 1) * CP_STRIDE + colw + m];
      acc = __builtin_amdgcn_wmma_f32_16x16x4_f32(false, a, false, ones,
                                                  (short)0, acc, false, false);
    }
  }
  // D[m][*] = colsum for column colbase+colw+m
  if ((lane & 15) == 0) {
    int half = lane >> 4;
    for (int r = 0; r < 8; ++r)
      partial[(size_t)blockIdx.y * N + colbase + colw + half * 8 + r] = acc[r];
  }
}

__global__ __launch_bounds__(256) void col_pass2(const float* __restrict__ partial,
                                                 float* __restrict__ v) {
  int j = blockIdx.x * 256 + threadIdx.x;
  float s = 0.0f;
  for (int rc = 0; rc < 64; ++rc) s += partial[(size_t)rc * N + j];  // fixed order
  v[j] = 1.0f / s;
}

__global__ __launch_bounds__(256) void logv_kernel(const float* __restrict__ v,
                                                   float* __restrict__ lv) {
  int j = blockIdx.x * 256 + threadIdx.x;
  lv[j] = logf(v[j]);
}

// ---------------------------------------------------------------------------
// Top-k per row: score = log(v[j]) - d[i][j]; radix-select 32nd largest,
// tie-break by lowest index (matches jax.lax.top_k); write 0/1 adjacency.
// ---------------------------------------------------------------------------
__global__ __launch_bounds__(256) void topk_kernel(const float* __restrict__ d,
                                                   const float* __restrict__ lv,
                                                   float* __restrict__ out) {
  __shared__ float sc[N];             // 32KB scores
  __shared__ unsigned hist[256];
  __shared__ unsigned s_prefix;
  __shared__ int s_rank;
  __shared__ int eqlist[64];
  __shared__ int eqcount;
  const int row = blockIdx.x;
  const int t = threadIdx.x;

  for (int k = 0; k < 32; ++k) {
    int j = k * 256 + t;
    sc[j] = lv[j] - d[(size_t)row * N + j];
  }
  if (t == 0) { s_prefix = 0u; s_rank = TOPK; eqcount = 0; }
  __syncthreads();

  const unsigned masks[4] = {0u, 0xFF000000u, 0xFFFF0000u, 0xFFFFFF00u};
  for (int round = 0; round < 4; ++round) {
    const int shift = 24 - round * 8;
    hist[t] = 0u;
    __syncthreads();
    const unsigned pfx = s_prefix;
    const unsigned pmask = masks[round];
    for (int k = 0; k < 32; ++k) {
      unsigned key = f2key(sc[k * 256 + t]);
      if ((key & pmask) == pfx) atomicAdd(&hist[(key >> shift) & 255], 1u);
    }
    __syncthreads();
    if (t == 0) {
      int r = s_rank;
      int b = 255;
      for (; b > 0; --b) {
        int h = (int)hist[b];
        if (r - h <= 0) break;
        r -= h;
      }
      s_rank = r;
      s_prefix = pfx | ((unsigned)b << shift);
    }
    __syncthreads();
  }
  const unsigned T = s_prefix;     // exact key of 32nd-largest
  const int need = s_rank;         // how many T-valued elements belong in top-32

  for (int k = 0; k < 32; ++k) {
    int j = k * 256 + t;
    unsigned key = f2key(sc[j]);
    float o = 0.0f;
    if (key > T) {
      o = 1.0f;
    } else if (key == T) {
      int pos = atomicAdd(&eqcount, 1);
      if (pos < 64) eqlist[pos] = j;
    }
    out[(size_t)row * N + j] = o;
  }
  __syncthreads();
  if (t == 0) {
    int n = eqcount < 64 ? eqcount : 64;
    for (int i = 1; i < n; ++i) {  // insertion sort by index (n is tiny)
      int x = eqlist[i], k2 = i - 1;
      while (k2 >= 0 && eqlist[k2] > x) { eqlist[k2 + 1] = eqlist[k2]; --k2; }
      eqlist[k2 + 1] = x;
    }
    int take = need < n ? need : n;
    for (int i = 0; i < take; ++i) out[(size_t)row * N + eqlist[i]] = 1.0f;
  }
}

extern "C" void kernel_launch(void* const* d_in, const int* in_sizes, int n_in,
                              void* d_out, int out_size, void* d_ws, size_t ws_size,
                              hipStream_t stream) {
  const float* d = (const float*)d_in[0];
  float* out = (float*)d_out;
  float* u = (float*)d_ws;        // 8192 f32
  float* v = u + N;               // 8192 f32
  float* lv = v + N;              // 8192 f32
  float* partial = lv + N;        // 64 * 8192 f32 (2 MB)

  init_v_kernel<<<N / 256, 256, 0, stream>>>(v);
  for (int it = 0; it < ITERS; ++it) {
    row_pass<<<N / 16, 256, 0, stream>>>(d, v, u);
    col_pass1<<<dim3(64, 64), 256, 0, stream>>>(d, u, partial);
    col_pass2<<<N / 256, 256, 0, stream>>>(partial, v);
  }
  logv_kernel<<<N / 256, 256, 0, stream>>>(v, lv);
  topk_kernel<<<N, 256, 0, stream>>>(d, lv, out);
}